// DifferentiableRobotModel_50938312130969
// MI455X (gfx1250) — compile-verified
//
#include <hip/hip_runtime.h>
#include <math.h>

// CDNA5 / gfx1250 ext-vector types for WMMA fragments
typedef __attribute__((ext_vector_type(2))) float v2f;   // A/B frag of V_WMMA_F32_16X16X4_F32 (16x4 f32 = 2 VGPRs)
typedef __attribute__((ext_vector_type(8))) float v8f;   // C/D frag (16x16 f32 = 8 VGPRs)

#define NLINK 16
#define NSPH  8
#define NTOT  128   // NLINK * NSPH

// One block per batch element. 128 threads = 4 wave32.
// FK chain -> world sphere centers -> pairwise penetration via
// V_WMMA_F32_16X16X4_F32 (d^2 = A*B + C with A=[-2x,-2y,-2z,1], B=[x,y,z,sq], C=sq_i)
// -> masked per-link max -> out[b][link].
__global__ __launch_bounds__(128)
void robot_selfcol_kernel(const float* __restrict__ q,             // [B,16]
                          const float* __restrict__ fixed_rot,     // [16,3,3]
                          const float* __restrict__ fixed_trans,   // [16,3]
                          const float* __restrict__ joint_axes,    // [16,3]
                          const float* __restrict__ link_spheres,  // [16,8,4]
                          const unsigned char* __restrict__ collision_mask, // [16,16] bool (1 byte)
                          float* __restrict__ out)                 // [B,16]
{
    __shared__ float s_frot[NLINK * 9];
    __shared__ float s_ftr[NLINK * 3];
    __shared__ float s_ax[NLINK * 3];
    __shared__ float s_sin[NLINK];
    __shared__ float s_cos[NLINK];
    __shared__ unsigned char s_mask[NLINK * NLINK];
    __shared__ float4 s_c[NTOT];   // world center x,y,z + |c|^2
    __shared__ float  s_r[NTOT];   // radius

    const int tid = threadIdx.x;
    const int b   = blockIdx.x;

    // ---- cooperative load of tiny constants into LDS ----
    if (tid < NLINK * 9) s_frot[tid] = fixed_rot[tid];
    if (tid < NLINK * 3) { s_ftr[tid] = fixed_trans[tid]; s_ax[tid] = joint_axes[tid]; }
    if (tid < NLINK) {
        float qv = q[b * NLINK + tid];
        s_sin[tid] = sinf(qv);
        s_cos[tid] = cosf(qv);
    }
    s_mask[tid]       = collision_mask[tid];
    s_mask[tid + 128] = collision_mask[tid + 128];
    __syncthreads();

    // ---- forward kinematics: each thread walks chain to its own link ----
    const int link = tid >> 3;               // 0..15
    float R00 = 1.f, R01 = 0.f, R02 = 0.f;
    float R10 = 0.f, R11 = 1.f, R12 = 0.f;
    float R20 = 0.f, R21 = 0.f, R22 = 1.f;
    float tx = 0.f, ty = 0.f, tz = 0.f;

    for (int k = 0; k <= link; ++k) {
        const float s  = s_sin[k], c = s_cos[k], oc = 1.0f - c;
        const float ax = s_ax[3 * k + 0], ay = s_ax[3 * k + 1], az = s_ax[3 * k + 2];
        // Rodrigues: J = c*I + s*[a]x + (1-c)*a a^T
        const float J00 = c + oc * ax * ax, J01 = oc * ax * ay - s * az, J02 = oc * ax * az + s * ay;
        const float J10 = oc * ax * ay + s * az, J11 = c + oc * ay * ay, J12 = oc * ay * az - s * ax;
        const float J20 = oc * ax * az - s * ay, J21 = oc * ay * az + s * ax, J22 = c + oc * az * az;
        const float* F = &s_frot[9 * k];
        // M = F @ J
        const float M00 = F[0]*J00 + F[1]*J10 + F[2]*J20;
        const float M01 = F[0]*J01 + F[1]*J11 + F[2]*J21;
        const float M02 = F[0]*J02 + F[1]*J12 + F[2]*J22;
        const float M10 = F[3]*J00 + F[4]*J10 + F[5]*J20;
        const float M11 = F[3]*J01 + F[4]*J11 + F[5]*J21;
        const float M12 = F[3]*J02 + F[4]*J12 + F[5]*J22;
        const float M20 = F[6]*J00 + F[7]*J10 + F[8]*J20;
        const float M21 = F[6]*J01 + F[7]*J11 + F[8]*J21;
        const float M22 = F[6]*J02 + F[7]*J12 + F[8]*J22;
        // t = t + R_parent @ ftrans  (uses R BEFORE update, matching reference)
        const float fx = s_ftr[3 * k + 0], fy = s_ftr[3 * k + 1], fz = s_ftr[3 * k + 2];
        tx += R00 * fx + R01 * fy + R02 * fz;
        ty += R10 * fx + R11 * fy + R12 * fz;
        tz += R20 * fx + R21 * fy + R22 * fz;
        // R = R @ M
        const float n00 = R00*M00 + R01*M10 + R02*M20;
        const float n01 = R00*M01 + R01*M11 + R02*M21;
        const float n02 = R00*M02 + R01*M12 + R02*M22;
        const float n10 = R10*M00 + R11*M10 + R12*M20;
        const float n11 = R10*M01 + R11*M11 + R12*M21;
        const float n12 = R10*M02 + R11*M12 + R12*M22;
        const float n20 = R20*M00 + R21*M10 + R22*M20;
        const float n21 = R20*M01 + R21*M11 + R22*M21;
        const float n22 = R20*M02 + R21*M12 + R22*M22;
        R00 = n00; R01 = n01; R02 = n02;
        R10 = n10; R11 = n11; R12 = n12;
        R20 = n20; R21 = n21; R22 = n22;
    }

    // ---- transform this thread's sphere to world frame ----
    {
        const float4 sp = ((const float4*)link_spheres)[tid];  // cx,cy,cz,r
        const float wx = R00 * sp.x + R01 * sp.y + R02 * sp.z + tx;
        const float wy = R10 * sp.x + R11 * sp.y + R12 * sp.z + ty;
        const float wz = R20 * sp.x + R21 * sp.y + R22 * sp.z + tz;
        const float sq = wx * wx + wy * wy + wz * wz;
        s_c[tid] = make_float4(wx, wy, wz, sq);
        s_r[tid] = sp.w;
    }
    __syncthreads();

    // ---- pairwise d^2 via WMMA: 8x8 grid of 16x16 tiles; wave w owns tile rows {w, w+4} ----
    const int lane = tid & 31;
    const int wv   = tid >> 5;
    const int half = lane >> 4;   // 0: lanes 0-15, 1: lanes 16-31
    const int ln   = lane & 15;

    for (int tr = 0; tr < 2; ++tr) {
        const int ti = wv + tr * 4;     // tile row 0..7

        // A fragment (16x4 f32): lanes 0-15 M=0..15; VGPR0 = K0 (lo) / K2 (hi), VGPR1 = K1 / K3.
        // A row m = [-2x, -2y, -2z, 1]
        const float4 ca = s_c[ti * 16 + ln];
        v2f a;
        if (half == 0) { a.x = -2.0f * ca.x; a.y = -2.0f * ca.y; }
        else           { a.x = -2.0f * ca.z; a.y = 1.0f;         }

        // C fragment: C_ij = sq_i  (VGPR v -> row v + half*8, same value across lanes of a half)
        v8f cacc;
        float rm[8];
        #pragma unroll
        for (int v = 0; v < 8; ++v) {
            const int m = ti * 16 + half * 8 + v;
            cacc[v] = s_c[m].w;
            rm[v]   = s_r[m];
        }

        const int li = ti * 2 + half;   // link index of this lane-half's rows

        // hoist the 8 collision-mask bytes for this lane's (li, lj(tj)) pairs into a bitmask
        unsigned int mbits = 0;
        #pragma unroll
        for (int tj = 0; tj < 8; ++tj) {
            const int lj = tj * 2 + (ln >> 3);
            mbits |= (s_mask[li * 16 + lj] ? 1u : 0u) << tj;
        }

        float run[8];
        #pragma unroll
        for (int v = 0; v < 8; ++v) run[v] = -100.0f;

        for (int tj = 0; tj < 8; ++tj) {
            // B fragment (4x16 f32), layout assumed symmetric to A:
            // lanes give N=0..15; VGPR0 = K0 (lo half) / K2 (hi half), VGPR1 = K1 / K3.
            // B col j = [x, y, z, sq_j]
            const int col = tj * 16 + ln;
            const float4 cb = s_c[col];
            v2f bf;
            if (half == 0) { bf.x = cb.x; bf.y = cb.y; }
            else           { bf.x = cb.z; bf.y = cb.w; }

            // D = A*B + C = (-2 c_i.c_j + sq_j) + sq_i = ||c_i - c_j||^2
            v8f d = __builtin_amdgcn_wmma_f32_16x16x4_f32(
                false, a, false, bf, (short)0, cacc, false, false);

            const bool ok = (mbits >> tj) & 1u;
            const float rn = s_r[col];
            #pragma unroll
            for (int v = 0; v < 8; ++v) {
                // raw v_sqrt_f32 (~1 ulp) instead of the IEEE-expanded sqrtf: input is
                // clamped to >=1e-12 and only feeds a max-reduction, so this is plenty.
                const float pen = rm[v] + rn - __builtin_amdgcn_sqrtf(fmaxf(d[v], 1e-12f));
                run[v] = fmaxf(run[v], ok ? pen : -100.0f);
            }
        }

        // reduce 8 row-values (all rows of link li for this lane) ...
        float lm = run[0];
        #pragma unroll
        for (int v = 1; v < 8; ++v) lm = fmaxf(lm, run[v]);
        // ... then max across the 16 lanes (columns) of this half
        #pragma unroll
        for (int off = 1; off < 16; off <<= 1)
            lm = fmaxf(lm, __shfl_xor(lm, off, 32));

        if (ln == 0) out[b * NLINK + li] = lm;   // lanes 0 and 16 write li and li+1's links
    }
}

extern "C" void kernel_launch(void* const* d_in, const int* in_sizes, int n_in,
                              void* d_out, int out_size, void* d_ws, size_t ws_size,
                              hipStream_t stream) {
    const float* q     = (const float*)d_in[0];
    const float* frot  = (const float*)d_in[1];
    const float* ftr   = (const float*)d_in[2];
    const float* axes  = (const float*)d_in[3];
    const float* sph   = (const float*)d_in[4];
    const unsigned char* mask = (const unsigned char*)d_in[5];  // jnp bool -> 1 byte/elem
    float* out = (float*)d_out;

    const int B = in_sizes[0] / NLINK;   // 8192
    robot_selfcol_kernel<<<B, 128, 0, stream>>>(q, frot, ftr, axes, sph, mask, out);
}